// RNNEncoder_32538672234954
// MI455X (gfx1250) — compile-verified
//
#include <hip/hip_runtime.h>

typedef __attribute__((ext_vector_type(2))) float v2f;
typedef __attribute__((ext_vector_type(4))) float v4f;
typedef __attribute__((ext_vector_type(8))) float v8f;
typedef int lru_v4i __attribute__((vector_size(16)));   // matches builtin param

namespace {
constexpr int kB = 16;
constexpr int kS = 2048;
constexpr int kI = 256;
constexpr int kH = 512;
constexpr int kM = kB * kS;             // 32768 GEMM rows
constexpr int kOutStride = 2 * kH;      // 1024 floats per (b,s) row: [re | im]
constexpr int kChunks = 64;             // scan chunks along S
constexpr int kChunkLen = kS / kChunks; // 32
constexpr int kNTile = 32;              // N columns per block (LDS-staged)
constexpr int kLdsFloats = kNTile * kI; // 8192 floats per matrix (32 KB)
}

#if __has_builtin(__builtin_amdgcn_global_load_async_to_lds_b128)
#define LRU_ASYNC_LDS 1
#else
#define LRU_ASYNC_LDS 0
#endif

__device__ __forceinline__ void lru_wait_async0() {
#if __has_builtin(__builtin_amdgcn_s_wait_asynccnt)
  __builtin_amdgcn_s_wait_asynccnt(0);
#else
  asm volatile("s_wait_asynccnt 0" ::: "memory");
#endif
}

__device__ __forceinline__ void lru_copy16_to_lds(const float* __restrict__ g,
                                                  float* __restrict__ l) {
#if LRU_ASYNC_LDS
  __builtin_amdgcn_global_load_async_to_lds_b128(
      (__attribute__((address_space(1))) lru_v4i*)(lru_v4i*)(void*)(float*)g,
      (__attribute__((address_space(3))) lru_v4i*)(lru_v4i*)(void*)l, 0, 0);
#else
  *(v4f*)l = *(const v4f*)g;
#endif
}

// ---------------------------------------------------------------------------
// Phase 1: Bx projection via fp32 WMMA, weights staged in LDS.
// Block = 8 waves = 8 M-tiles (16 rows each) x one 32-column N-tile.
// LDS holds W_re/W_im chunk [32 cols x 256 K] = 64 KB, filled once per block
// with async global->LDS DMA, then read as the WMMA B operand via ds loads.
//   A (16x4 f32): lane l<16 holds K=0,1 for row m0+l; lane l+16 holds K=2,3.
//   B (4x16 f32): lane l<16 holds K=0,1 for col n;    lane l+16 holds K=2,3.
//   C/D (16x16 f32): VGPR r = row m0+r (lanes 0-15) / m0+8+r (lanes 16-31).
// ---------------------------------------------------------------------------
__global__ __launch_bounds__(256)
void lru_proj_wmma(const float* __restrict__ x,
                   const float* __restrict__ w_re,
                   const float* __restrict__ w_im,
                   float* __restrict__ out)
{
  __shared__ float smem[2 * kLdsFloats];   // [0..8191]=W_re, [8192..]=W_im

  const int ntile  = blockIdx.x & 15;      // 16 N-tiles of 32 cols (H = 512)
  const int msuper = blockIdx.x >> 4;      // 256 super-tiles of 8 M-tiles
  const int waveid = threadIdx.x >> 5;
  const int lane   = threadIdx.x & 31;
  const int n0 = ntile * kNTile;
  const int m0 = (msuper * 8 + waveid) * 16;
  const int lm = lane & 15;                // A: row-in-tile, B: col-in-tile
  const int kh = lane >> 4;                // selects K pair {0,1} vs {2,3}

  // ---- cooperative LDS fill: 32 contiguous rows of each weight matrix ----
  {
    const float* src_re = w_re + (size_t)n0 * kI;  // 32*256 floats contiguous
    const float* src_im = w_im + (size_t)n0 * kI;
    for (int i = threadIdx.x; i < kLdsFloats / 4; i += 256) {
      lru_copy16_to_lds(src_re + 4 * i, smem + 4 * i);
      lru_copy16_to_lds(src_im + 4 * i, smem + kLdsFloats + 4 * i);
    }
#if LRU_ASYNC_LDS
    lru_wait_async0();
#endif
    __syncthreads();
  }

  const float* arow    = x + (size_t)(m0 + lm) * kI + 2 * kh;
  const float* lds_re0 = smem + (lm * kI + 2 * kh);
  const float* lds_im0 = smem + kLdsFloats + (lm * kI + 2 * kh);

  v8f accR[2] = {};
  v8f accI[2] = {};

  for (int k0 = 0; k0 < kI; k0 += 4) {
    const v2f a = *(const v2f*)(arow + k0);
#pragma unroll
    for (int j = 0; j < 2; ++j) {
      const v2f br = *(const v2f*)(lds_re0 + j * 16 * kI + k0);
      const v2f bi = *(const v2f*)(lds_im0 + j * 16 * kI + k0);
      // 8 args: (neg_a, A, neg_b, B, c_mod, C, reuse_a, reuse_b)
      accR[j] = __builtin_amdgcn_wmma_f32_16x16x4_f32(
          false, a, false, br, (short)0, accR[j], false, false);
      accI[j] = __builtin_amdgcn_wmma_f32_16x16x4_f32(
          false, a, false, bi, (short)0, accI[j], false, false);
    }
  }

#pragma unroll
  for (int j = 0; j < 2; ++j) {
    const int n = n0 + j * 16 + lm;
#pragma unroll
    for (int r = 0; r < 8; ++r) {
      const size_t row = (size_t)(m0 + r + 8 * kh);
      out[row * kOutStride + n]      = accR[j][r];
      out[row * kOutStride + kH + n] = accI[j][r];
    }
  }
}

// ---------------------------------------------------------------------------
// Phase 2a: chunk-local scans with zero initial state.
// tid = (b << 15) | (c << 9) | h  -> 524288 threads, h fastest (coalesced).
// ---------------------------------------------------------------------------
__global__ __launch_bounds__(256)
void lru_scan_local(const float* __restrict__ out,
                    const float* __restrict__ nu,
                    const float* __restrict__ th,
                    float* __restrict__ f_re,
                    float* __restrict__ f_im)
{
  const int tid = blockIdx.x * blockDim.x + threadIdx.x;
  const int h = tid & (kH - 1);
  const int c = (tid >> 9) & (kChunks - 1);
  const int b = tid >> 15;

  const float mag = expf(-expf(nu[h]));
  const float lr = mag * cosf(th[h]);
  const float li = mag * sinf(th[h]);

  const float* row = out + (size_t)(b * kS + c * kChunkLen) * kOutStride + h;
  float hr = 0.f, hi = 0.f;
  for (int s = 0; s < kChunkLen; ++s) {
    const float br = row[0];
    const float bi = row[kH];
    const float nr = fmaf(lr, hr, fmaf(-li, hi, br));
    const float ni = fmaf(lr, hi, fmaf(li, hr, bi));
    hr = nr; hi = ni;
    row += kOutStride;
  }
  f_re[tid] = hr;
  f_im[tid] = hi;
}

// ---------------------------------------------------------------------------
// Phase 2b: exclusive scan over chunk finals -> chunk initial states, with
// lam^L = exp(-L*exp(nu)) * (cos(L*theta) + i sin(L*theta)). In place.
// ---------------------------------------------------------------------------
__global__ __launch_bounds__(256)
void lru_scan_combine(const float* __restrict__ nu,
                      const float* __restrict__ th,
                      float* __restrict__ f_re,
                      float* __restrict__ f_im)
{
  const int tid = blockIdx.x * blockDim.x + threadIdx.x;
  const int h = tid & (kH - 1);
  const int b = tid >> 9;

  const float magL = expf(-(float)kChunkLen * expf(nu[h]));
  const float angL = (float)kChunkLen * th[h];
  const float Lr = magL * cosf(angL);
  const float Li = magL * sinf(angL);

  float sr = 0.f, si = 0.f;
  for (int c = 0; c < kChunks; ++c) {
    const size_t idx = (size_t)(b * kChunks + c) * kH + h;
    const float fr = f_re[idx];
    const float fi = f_im[idx];
    f_re[idx] = sr;              // overwrite final with start state
    f_im[idx] = si;
    const float nr = fmaf(Lr, sr, fmaf(-Li, si, fr));
    const float ni = fmaf(Lr, si, fmaf(Li, sr, fi));
    sr = nr; si = ni;
  }
}

// ---------------------------------------------------------------------------
// Phase 2c: re-scan each chunk from its correct initial state, writing the
// scanned states in place over d_out.
// ---------------------------------------------------------------------------
__global__ __launch_bounds__(256)
void lru_scan_final(const float* __restrict__ nu,
                    const float* __restrict__ th,
                    const float* __restrict__ f_re,
                    const float* __restrict__ f_im,
                    float* __restrict__ out)
{
  const int tid = blockIdx.x * blockDim.x + threadIdx.x;
  const int h = tid & (kH - 1);
  const int c = (tid >> 9) & (kChunks - 1);
  const int b = tid >> 15;

  const float mag = expf(-expf(nu[h]));
  const float lr = mag * cosf(th[h]);
  const float li = mag * sinf(th[h]);

  float* row = out + (size_t)(b * kS + c * kChunkLen) * kOutStride + h;
  float hr = f_re[tid];
  float hi = f_im[tid];
  for (int s = 0; s < kChunkLen; ++s) {
    const float br = row[0];
    const float bi = row[kH];
    const float nr = fmaf(lr, hr, fmaf(-li, hi, br));
    const float ni = fmaf(lr, hi, fmaf(li, hr, bi));
    hr = nr; hi = ni;
    row[0]  = hr;
    row[kH] = hi;
    row += kOutStride;
  }
}

extern "C" void kernel_launch(void* const* d_in, const int* in_sizes, int n_in,
                              void* d_out, int out_size, void* d_ws, size_t ws_size,
                              hipStream_t stream) {
  const float* x    = (const float*)d_in[0];  // [B,S,I]
  const float* nu   = (const float*)d_in[1];  // [H]
  const float* th   = (const float*)d_in[2];  // [H]
  const float* w_re = (const float*)d_in[3];  // [H,I]
  const float* w_im = (const float*)d_in[4];  // [H,I]
  float* out = (float*)d_out;                 // [B,S,2H]

  float* f_re = (float*)d_ws;                     // [B*Chunks*H]
  float* f_im = f_re + (size_t)kB * kChunks * kH; // [B*Chunks*H]

  // Phase 1: 256 M-superblocks x 16 N-tiles = 4096 blocks of 8 waves.
  lru_proj_wmma<<<(kM / (16 * 8)) * (kH / kNTile), 256, 0, stream>>>(x, w_re, w_im, out);

  // Phase 2: chunked parallel scan, in place over d_out.
  lru_scan_local  <<<(kB * kChunks * kH) / 256, 256, 0, stream>>>(out, nu, th, f_re, f_im);
  lru_scan_combine<<<(kB * kH) / 256,           256, 0, stream>>>(nu, th, f_re, f_im);
  lru_scan_final  <<<(kB * kChunks * kH) / 256, 256, 0, stream>>>(nu, th, f_re, f_im, out);
}